// QRNNModel_8315056685299
// MI455X (gfx1250) — compile-verified
//
#include <hip/hip_runtime.h>
#include <hip/hip_bf16.h>

#define NQ    12
#define ND    4
#define TSTEP 8
#define BATCH 1024
#define DIM   4096   // 2^NQ

typedef float v2f __attribute__((ext_vector_type(2)));
typedef float v8f __attribute__((ext_vector_type(8)));

// ---------------- precomputed gate data (rebuilt deterministically each launch) ----
__device__ float g_gAr[ND][2][2];          // data-qubit combined RX*RZ*RX (real)
__device__ float g_gAi[ND][2][2];          // (imag)
__device__ float g_U1r[256], g_U1i[256];   // kron(qubits 4..7)  row-major [r][c]
__device__ float g_U2Tr[256], g_U2Ti[256]; // kron(qubits 8..11) transposed: [k][c] = U2[c][k]
__device__ float g_Dr[DIM], g_Di[DIM];     // entangling-ring diagonal phases

// ---------------- complex 2x2 multiply ----------------
__device__ __forceinline__ void cmul2x2(const float ar[2][2], const float ai[2][2],
                                        const float br[2][2], const float bi[2][2],
                                        float cr[2][2], float ci[2][2]) {
#pragma unroll
  for (int r = 0; r < 2; ++r)
#pragma unroll
    for (int c = 0; c < 2; ++c) {
      float sr = 0.f, si = 0.f;
#pragma unroll
      for (int k = 0; k < 2; ++k) {
        sr += ar[r][k] * br[k][c] - ai[r][k] * bi[k][c];
        si += ar[r][k] * bi[k][c] + ai[r][k] * br[k][c];
      }
      cr[r][c] = sr; ci[r][c] = si;
    }
}

// ---------------- setup kernel: build gate tables from weights ----------------
__global__ __launch_bounds__(256) void qrnn_setup(const float* __restrict__ w /*12x4*/) {
  __shared__ float Ar[NQ][2][2], Ai[NQ][2][2];
  const int tid = threadIdx.x;

  if (tid < NQ) {
    float a0 = w[tid * 4 + 0], a1 = w[tid * 4 + 1], a2 = w[tid * 4 + 2];
    float c0 = cosf(a0 * 0.5f), s0 = sinf(a0 * 0.5f);
    float c2 = cosf(a2 * 0.5f), s2 = sinf(a2 * 0.5f);
    float zc = cosf(a1 * 0.5f), zs = sinf(a1 * 0.5f);
    // RX(a) = [[c, -i s],[-i s, c]];  RZ(p) = diag(e^{-ip/2}, e^{+ip/2})
    float X0r[2][2] = {{c0, 0.f}, {0.f, c0}};
    float X0i[2][2] = {{0.f, -s0}, {-s0, 0.f}};
    float Zr[2][2]  = {{zc, 0.f}, {0.f, zc}};
    float Zi[2][2]  = {{-zs, 0.f}, {0.f, zs}};
    float X2r[2][2] = {{c2, 0.f}, {0.f, c2}};
    float X2i[2][2] = {{0.f, -s2}, {-s2, 0.f}};
    float M1r[2][2], M1i[2][2], Mr[2][2], Mi[2][2];
    // gates applied in order RX(w0), RZ(w1), RX(w2)  ->  M = RX(w2)*RZ(w1)*RX(w0)
    cmul2x2(Zr, Zi, X0r, X0i, M1r, M1i);
    cmul2x2(X2r, X2i, M1r, M1i, Mr, Mi);
#pragma unroll
    for (int r = 0; r < 2; ++r)
#pragma unroll
      for (int c = 0; c < 2; ++c) { Ar[tid][r][c] = Mr[r][c]; Ai[tid][r][c] = Mi[r][c]; }
  }
  __syncthreads();

  if (tid < ND * 4) { // export data-qubit gates
    int j = tid >> 2, r = (tid >> 1) & 1, c = tid & 1;
    g_gAr[j][r][c] = Ar[j][r][c];
    g_gAi[j][r][c] = Ai[j][r][c];
  }

  { // Kronecker products for qubit groups {4..7} and {8..11}
    int r = tid >> 4, c = tid & 15;
    float pr = 1.f, pi = 0.f;
#pragma unroll
    for (int j = 0; j < 4; ++j) { // qubit 4+j : bit (3-j) of group index
      int rb = (r >> (3 - j)) & 1, cb = (c >> (3 - j)) & 1;
      float mr = Ar[4 + j][rb][cb], mi = Ai[4 + j][rb][cb];
      float nr = pr * mr - pi * mi, ni = pr * mi + pi * mr;
      pr = nr; pi = ni;
    }
    g_U1r[tid] = pr; g_U1i[tid] = pi;

    // U2T[k][c] = U2[c][k]
    int kk = r, cc = c;
    pr = 1.f; pi = 0.f;
#pragma unroll
    for (int j = 0; j < 4; ++j) {
      int rb = (cc >> (3 - j)) & 1, cb = (kk >> (3 - j)) & 1;
      float mr = Ar[8 + j][rb][cb], mi = Ai[8 + j][rb][cb];
      float nr = pr * mr - pi * mi, ni = pr * mi + pi * mr;
      pr = nr; pi = ni;
    }
    g_U2Tr[tid] = pr; g_U2Ti[tid] = pi;
  }

  // diagonal of the CNOT-RZ-CNOT ring: phase e^{-i p/2} if bit_c==bit_t else e^{+i p/2}
  for (int n = tid; n < DIM; n += 256) {
    float s = 0.f;
#pragma unroll
    for (int j = 0; j < NQ - 1; ++j) {
      int bc = (n >> (NQ - 1 - j)) & 1;
      int bt = (n >> (NQ - 2 - j)) & 1;
      float phi = w[(j + 1) * 4 + 3];
      s += (bc == bt) ? -0.5f * phi : 0.5f * phi;
    }
    int bc = n & 1;              // qubit 11
    int bt = (n >> (NQ - 1)) & 1; // qubit 0
    float phi = w[3];
    s += (bc == bt) ? -0.5f * phi : 0.5f * phi;
    g_Dr[n] = cosf(s);
    g_Di[n] = sinf(s);
  }
}

// ---------------- f32 WMMA wrapper ----------------
__device__ __forceinline__ v8f wmma4(v2f a, v2f b, v8f c) {
  // (neg_a, A, neg_b, B, c_mod, C, reuse_a, reuse_b)
  return __builtin_amdgcn_wmma_f32_16x16x4_f32(false, a, false, b, (short)0, c, false, false);
}

// D = G * S_tile  (complex 16x16 * 16x16), in-place on the state tile.
// G: row-major 16x16 (real/imag).  State tile element [k][n] = S[base + k*rstride + n].
__device__ __forceinline__ void mm_left(const float* __restrict__ Gr, const float* __restrict__ Gi,
                                        float* Sr, float* Si,
                                        int base, int rstride, int h, int l16) {
  v2f ar[4], ai[4], br[4], bi[4];
#pragma unroll
  for (int k = 0; k < 4; ++k) {
    int k0 = k * 4 + 2 * h;                 // A/B K index = 4k + v + 2*(lane>=16)
    ar[k].x = Gr[l16 * 16 + k0];     ar[k].y = Gr[l16 * 16 + k0 + 1];
    ai[k].x = Gi[l16 * 16 + k0];     ai[k].y = Gi[l16 * 16 + k0 + 1];
    br[k].x = Sr[base + k0 * rstride + l16];       br[k].y = Sr[base + (k0 + 1) * rstride + l16];
    bi[k].x = Si[base + k0 * rstride + l16];       bi[k].y = Si[base + (k0 + 1) * rstride + l16];
  }
  v8f aRR = {}, aII = {}, aI = {};
#pragma unroll
  for (int k = 0; k < 4; ++k) {
    aRR = wmma4(ar[k], br[k], aRR);   // Re*Re
    aII = wmma4(ai[k], bi[k], aII);   // Im*Im
    aI  = wmma4(ar[k], bi[k], aI);    // Re*Im
    aI  = wmma4(ai[k], br[k], aI);    // Im*Re
  }
#pragma unroll
  for (int v = 0; v < 8; ++v) {
    int row = v + 8 * h;               // C/D row = vgpr + 8*(lane>=16)
    Sr[base + row * rstride + l16] = aRR[v] - aII[v];
    Si[base + row * rstride + l16] = aI[v];
  }
}

// D = S_tile * B  with fused diagonal-phase multiply on write-back (ring gate).
// A = state tile [m][k] = S[base + m*16 + k];  B = U2T row-major [k][n].
__device__ __forceinline__ void mm_right_diag(float* Sr, float* Si, int base,
                                              const float* __restrict__ BTr,
                                              const float* __restrict__ BTi,
                                              int h, int l16) {
  v2f ar[4], ai[4], br[4], bi[4];
#pragma unroll
  for (int k = 0; k < 4; ++k) {
    int k0 = k * 4 + 2 * h;
    ar[k].x = Sr[base + l16 * 16 + k0];  ar[k].y = Sr[base + l16 * 16 + k0 + 1];
    ai[k].x = Si[base + l16 * 16 + k0];  ai[k].y = Si[base + l16 * 16 + k0 + 1];
    br[k].x = BTr[k0 * 16 + l16];        br[k].y = BTr[(k0 + 1) * 16 + l16];
    bi[k].x = BTi[k0 * 16 + l16];        bi[k].y = BTi[(k0 + 1) * 16 + l16];
  }
  v8f aRR = {}, aII = {}, aI = {};
#pragma unroll
  for (int k = 0; k < 4; ++k) {
    aRR = wmma4(ar[k], br[k], aRR);
    aII = wmma4(ai[k], bi[k], aII);
    aI  = wmma4(ar[k], bi[k], aI);
    aI  = wmma4(ai[k], br[k], aI);
  }
#pragma unroll
  for (int v = 0; v < 8; ++v) {
    int row = v + 8 * h;
    int idx = base + row * 16 + l16;
    float vr = aRR[v] - aII[v];
    float vi = aI[v];
    float dr = g_Dr[idx], di = g_Di[idx];
    Sr[idx] = vr * dr - vi * di;
    Si[idx] = vr * di + vi * dr;
  }
}

// ---------------- main kernel: one workgroup per batch element ----------------
__global__ __launch_bounds__(256) void qrnn_main(const float* __restrict__ inputs,
                                                 const float* __restrict__ W_out,
                                                 const float* __restrict__ b_out,
                                                 float* __restrict__ out) {
  __shared__ float Sr[DIM], Si[DIM];     // 32 KB resident state
  __shared__ float U0r[256], U0i[256];   // per-step batched group-0 gate
  __shared__ float red[256];

  const int tid  = threadIdx.x;
  const int b    = blockIdx.x;
  const int lane = tid & 31;
  const int wave = tid >> 5;             // 8 waves (wave32)
  const int h    = lane >> 4;
  const int l16  = lane & 15;

  for (int i = tid; i < DIM; i += 256) { Sr[i] = 0.f; Si[i] = 0.f; }
  if (tid == 0) Sr[0] = 1.f;             // |0...0>
  __syncthreads();

  for (int t = 0; t < TSTEP; ++t) {
    // ---- batched group-0 gate: kron_j ( A_j * RY(arccos(x)) ), qubits 0..3 ----
    float x  = inputs[b * TSTEP + t];
    float cc = sqrtf(fmaxf(0.f, (1.f + x) * 0.5f));  // cos(arccos(x)/2)
    float ss = sqrtf(fmaxf(0.f, (1.f - x) * 0.5f));  // sin(arccos(x)/2)
    {
      int r = tid >> 4, c = tid & 15;
      float pr = 1.f, pi = 0.f;
#pragma unroll
      for (int j = 0; j < ND; ++j) {
        int rb = (r >> (3 - j)) & 1, cb = (c >> (3 - j)) & 1;
        float a0r = g_gAr[j][rb][0], a0i = g_gAi[j][rb][0];
        float a1r = g_gAr[j][rb][1], a1i = g_gAi[j][rb][1];
        // G = A * RY: col0 = A[:,0]*c + A[:,1]*s ; col1 = A[:,1]*c - A[:,0]*s
        float gr = (cb == 0) ? (a0r * cc + a1r * ss) : (a1r * cc - a0r * ss);
        float gi = (cb == 0) ? (a0i * cc + a1i * ss) : (a1i * cc - a0i * ss);
        float nr = pr * gr - pi * gi, ni = pr * gi + pi * gr;
        pr = nr; pi = ni;
      }
      U0r[tid] = pr; U0i[tid] = pi;
    }
    __syncthreads();

    // ---- phase 1: qubits 0..3 : U0 x State(16 x 256); 16 column tiles / 8 waves
#pragma unroll
    for (int tt = 0; tt < 2; ++tt)
      mm_left(&U0r[0], &U0i[0], Sr, Si, (wave * 2 + tt) * 16, 256, h, l16);
    __syncthreads();

    // ---- phase 2: qubits 4..7 : 16 independent 16x16 slices (stride 16)
#pragma unroll
    for (int tt = 0; tt < 2; ++tt)
      mm_left(g_U1r, g_U1i, Sr, Si, (wave * 2 + tt) * 256, 16, h, l16);
    __syncthreads();

    // ---- phase 3: qubits 8..11 : right-multiply by U2^T, fused ring diagonal
#pragma unroll
    for (int tt = 0; tt < 2; ++tt)
      mm_right_diag(Sr, Si, (wave * 2 + tt) * 256, g_U2Tr, g_U2Ti, h, l16);
    __syncthreads();
  }

  // ---- readout: sum_n |psi_n|^2 * sum_j W_j*(1-2*bit_j(n)), j over data qubits
  float w0 = W_out[0], w1 = W_out[1], w2 = W_out[2], w3 = W_out[3];
  float partial = 0.f;
  for (int i = tid; i < DIM; i += 256) {
    float p = Sr[i] * Sr[i] + Si[i] * Si[i];
    float wz = (((i >> 11) & 1) ? -w0 : w0) + (((i >> 10) & 1) ? -w1 : w1)
             + (((i >> 9) & 1) ? -w2 : w2) + (((i >> 8) & 1) ? -w3 : w3);
    partial += p * wz;
  }
  red[tid] = partial;
  __syncthreads();
  for (int s = 128; s > 0; s >>= 1) {
    if (tid < s) red[tid] += red[tid + s];
    __syncthreads();
  }
  if (tid == 0) out[b] = red[0] + b_out[0];
}

extern "C" void kernel_launch(void* const* d_in, const int* in_sizes, int n_in,
                              void* d_out, int out_size, void* d_ws, size_t ws_size,
                              hipStream_t stream) {
  const float* inputs  = (const float*)d_in[0]; // (1024, 8)
  const float* weights = (const float*)d_in[1]; // (12, 4)
  const float* W_out   = (const float*)d_in[2]; // (1, 4)
  const float* b_out   = (const float*)d_in[3]; // (1,)
  float* out = (float*)d_out;                   // (1024, 1)
  (void)in_sizes; (void)n_in; (void)out_size; (void)d_ws; (void)ws_size;

  hipLaunchKernelGGL(qrnn_setup, dim3(1), dim3(256), 0, stream, weights);
  hipLaunchKernelGGL(qrnn_main, dim3(BATCH), dim3(256), 0, stream,
                     inputs, W_out, b_out, out);
}